// RefractiveSurface_43851616092400
// MI455X (gfx1250) — compile-verified
//
#include <hip/hip_runtime.h>
#include <cstdint>

typedef float v4f __attribute__((ext_vector_type(4)));

#define TPB 256
#define RAYS_PER_THREAD 4
#define TILE_RAYS (TPB * RAYS_PER_THREAD)   // 1024 rays per tile
#define NBLK 2048

// ---------------------------------------------------------------------------
// Single-ray trace: sphere intersection + aperture clip + Snell refraction.
// Writes 6 floats (hit point, refracted dir), zeroed if invalid.
// ---------------------------------------------------------------------------
__device__ __forceinline__ void trace1(float px, float py, float pz,
                                       float vx, float vy, float vz,
                                       float R0, float cx, float invR, float eta,
                                       float* __restrict__ o) {
    // PC = P - C  (C = (Z0 + R, 0, 0))
    float ex = px - cx, ey = py, ez = pz;
    float b = 2.0f * (vx * ex + vy * ey + vz * ez);
    float c = ex * ex + ey * ey + ez * ez - R0 * R0;
    float disc = b * b - 4.0f * c;
    float sd = __builtin_amdgcn_sqrtf(fmaxf(disc, 0.0f));
    float t1 = (-b - sd) * 0.5f;
    float t2 = (-b + sd) * 0.5f;
    float tt = (t1 > 1e-6f) ? t1 : t2;
    bool hit = (disc >= 0.0f) && (tt > 1e-6f);
    float qx = fmaf(tt, vx, px);
    float qy = fmaf(tt, vy, py);
    float qz = fmaf(tt, vz, pz);
    float r2 = qy * qy + qz * qz;
    hit = hit && (r2 <= 400.0f);          // (DIAMETER/2)^2 = 20^2
    // surface normal, oriented against the ray
    float nx = (qx - cx) * invR, ny = qy * invR, nz = qz * invR;
    float ndv = nx * vx + ny * vy + nz * vz;
    if (ndv > 0.0f) { nx = -nx; ny = -ny; nz = -nz; }
    float cosi = -(nx * vx + ny * vy + nz * vz);
    float sin2t = eta * eta * (1.0f - cosi * cosi);
    bool ok = sin2t <= 1.0f;
    float cost = __builtin_amdgcn_sqrtf(fmaxf(1.0f - sin2t, 0.0f));
    float kf = eta * cosi - cost;
    float tx = fmaf(eta, vx, kf * nx);
    float ty = fmaf(eta, vy, kf * ny);
    float tz = fmaf(eta, vz, kf * nz);
    bool val = hit && ok;
    o[0] = val ? qx : 0.0f;
    o[1] = val ? qy : 0.0f;
    o[2] = val ? qz : 0.0f;
    o[3] = val ? tx : 0.0f;
    o[4] = val ? ty : 0.0f;
    o[5] = val ? tz : 0.0f;
}

// ---------------------------------------------------------------------------
// Persistent-grid streaming kernel with async global->LDS double buffering.
// Each thread handles 4 rays/tile; each thread consumes exactly the LDS
// region it DMA'd, so only per-wave s_wait_asynccnt is needed (no barriers).
// ---------------------------------------------------------------------------
__global__ __launch_bounds__(TPB) void refract_async_kernel(
    const float* __restrict__ P, const float* __restrict__ V,
    const float* __restrict__ Rp, const float* __restrict__ n1p,
    const float* __restrict__ n2p, float* __restrict__ out, int nrays)
{
    // [buffer][chunk][thread]: chunks 0..2 = P (12 floats), 3..5 = V (12 floats)
    __shared__ v4f smem[2][6][TPB];

    const uint32_t t = threadIdx.x;
    // Low 32 bits of the generic LDS address = wave-relative LDS byte offset.
    const uint32_t lbase = (uint32_t)(uintptr_t)(&smem[0][0][0]);
    const uint64_t pB = (uint64_t)(uintptr_t)P;
    const uint64_t vB = (uint64_t)(uintptr_t)V;
    const int numTiles = nrays / TILE_RAYS;

    const float R0  = Rp[0];
    const float eta = n1p[0] / n2p[0];
    const float cx  = 10.0f + R0;        // Z0 + R
    const float invR = 1.0f / R0;

    // Issue 6 async b128 DMA loads (48B of P + 48B of V for this thread's rays).
    auto issue = [&](uint32_t buf, uint32_t tile) {
        uint32_t gb = tile * 12288u + t * 48u;                 // global byte offset
        uint32_t lb = lbase + (buf * 6u * TPB + t) * 16u;      // LDS byte offset
#pragma unroll
        for (uint32_t k = 0; k < 3; ++k) {
            uint32_t lp = lb + k * (TPB * 16u);
            uint32_t lv = lb + (k + 3u) * (TPB * 16u);
            uint32_t gp = gb + k * 16u;
            asm volatile("global_load_async_to_lds_b128 %0, %1, %2 th:TH_LOAD_NT"
                         :: "v"(lp), "v"(gp), "s"(pB) : "memory");
            asm volatile("global_load_async_to_lds_b128 %0, %1, %2 th:TH_LOAD_NT"
                         :: "v"(lv), "v"(gp), "s"(vB) : "memory");
        }
    };

    int buf = 0;
    int tile = (int)blockIdx.x;
    if (tile < numTiles) issue(0u, (uint32_t)tile);

    for (; tile < numTiles; tile += (int)gridDim.x) {
        int nxt = tile + (int)gridDim.x;
        bool more = (nxt < numTiles);
        if (more) {
            issue((uint32_t)(buf ^ 1), (uint32_t)nxt);
            asm volatile("s_wait_asynccnt 6" ::: "memory");  // current tile's 6 done
        } else {
            asm volatile("s_wait_asynccnt 0" ::: "memory");
        }

        v4f cP0 = smem[buf][0][t], cP1 = smem[buf][1][t], cP2 = smem[buf][2][t];
        v4f cV0 = smem[buf][3][t], cV1 = smem[buf][4][t], cV2 = smem[buf][5][t];
        float pf[12] = {cP0[0], cP0[1], cP0[2], cP0[3],
                        cP1[0], cP1[1], cP1[2], cP1[3],
                        cP2[0], cP2[1], cP2[2], cP2[3]};
        float vf[12] = {cV0[0], cV0[1], cV0[2], cV0[3],
                        cV1[0], cV1[1], cV1[2], cV1[3],
                        cV2[0], cV2[1], cV2[2], cV2[3]};

        float of[24];
#pragma unroll
        for (int j = 0; j < RAYS_PER_THREAD; ++j) {
            trace1(pf[3 * j], pf[3 * j + 1], pf[3 * j + 2],
                   vf[3 * j], vf[3 * j + 1], vf[3 * j + 2],
                   R0, cx, invR, eta, &of[6 * j]);
        }

        // 96 contiguous, 16B-aligned bytes of output -> 6 NT b128 stores.
        size_t obase = (size_t)tile * (TILE_RAYS * 6) + (size_t)t * 24u;
        v4f* op = (v4f*)(out + obase);
        v4f o0 = {of[0],  of[1],  of[2],  of[3]};
        v4f o1 = {of[4],  of[5],  of[6],  of[7]};
        v4f o2 = {of[8],  of[9],  of[10], of[11]};
        v4f o3 = {of[12], of[13], of[14], of[15]};
        v4f o4 = {of[16], of[17], of[18], of[19]};
        v4f o5 = {of[20], of[21], of[22], of[23]};
        __builtin_nontemporal_store(o0, op + 0);
        __builtin_nontemporal_store(o1, op + 1);
        __builtin_nontemporal_store(o2, op + 2);
        __builtin_nontemporal_store(o3, op + 3);
        __builtin_nontemporal_store(o4, op + 4);
        __builtin_nontemporal_store(o5, op + 5);

        buf ^= 1;
    }

    // Tail (rays beyond the last full tile) — direct path, no LDS.
    int rem = nrays - numTiles * TILE_RAYS;
    if (rem > 0) {
        int gid = (int)blockIdx.x * TPB + (int)t;
        if (gid < rem) {
            int r = numTiles * TILE_RAYS + gid;
            const float* p = P + (size_t)r * 3;
            const float* v = V + (size_t)r * 3;
            float o[6];
            trace1(p[0], p[1], p[2], v[0], v[1], v[2], R0, cx, invR, eta, o);
            float* ob = out + (size_t)r * 6;
#pragma unroll
            for (int i = 0; i < 6; ++i) ob[i] = o[i];
        }
    }
}

extern "C" void kernel_launch(void* const* d_in, const int* in_sizes, int n_in,
                              void* d_out, int out_size, void* d_ws, size_t ws_size,
                              hipStream_t stream) {
    const float* P  = (const float*)d_in[0];
    const float* V  = (const float*)d_in[1];
    const float* R  = (const float*)d_in[2];
    const float* n1 = (const float*)d_in[3];
    const float* n2 = (const float*)d_in[4];
    float* out = (float*)d_out;

    int nrays = in_sizes[0] / 3;
    int numTiles = nrays / TILE_RAYS;

    int blocks = NBLK;
    if (numTiles < blocks) blocks = (numTiles > 0) ? numTiles : 1;
    int rem = nrays - numTiles * TILE_RAYS;
    int tailBlocks = (rem + TPB - 1) / TPB;
    if (blocks < tailBlocks) blocks = tailBlocks;

    refract_async_kernel<<<blocks, TPB, 0, stream>>>(P, V, R, n1, n2, out, nrays);
}